// SpectrumRefinerLoss_13640816132153
// MI455X (gfx1250) — compile-verified
//
#include <hip/hip_runtime.h>
#include <hip/hip_bf16.h>
#include <math.h>

// Problem constants (match reference)
#define B_DIM   64
#define KMAX    8
#define G_PHI   512
#define G_THETA 256
#define ROWS    8                              // phi rows per block
#define NBLK    (B_DIM * (G_PHI / ROWS))       // 4096 blocks -> 16KB of partials in d_ws
#define EPS_F   1e-7f
#define F_ALPHA 0.25f

typedef __attribute__((ext_vector_type(2))) float v2f;
typedef __attribute__((ext_vector_type(8))) float v8f;

// Wave32 sum via two chained V_WMMA_F32_16X16X4_F32, no cross-lane shuffles.
//
// WMMA #1: A(16x4) with lane L<16 -> A[L][0]=x, lane L>=16 -> A[L-16][2]=x
// (K=1,3 zero); B = ones (layout-independent). D[m][n] = x[m] + x[m+16].
// Per-lane sum s of the 8 D regs: lanes 0-15 hold S_A = sum over rows 0-7,
// lanes 16-31 hold S_B = sum over rows 8-15 (columns are identical).
//
// WMMA #2: feed s back the same way: D2[m][n] = s[m] + s[m+16] = S_A + S_B
// = full 32-lane total, replicated to EVERY lane/register. Result = d2[0].
//
// EXEC must be all ones at the call site (WMMA restriction).
__device__ __forceinline__ float wave_sum32(float x) {
#if __has_builtin(__builtin_amdgcn_wmma_f32_16x16x4_f32)
  v2f bo; bo[0] = 1.0f; bo[1] = 1.0f;
  v8f c = {};
  v2f a;  a[0] = x; a[1] = 0.0f;
  v8f d = __builtin_amdgcn_wmma_f32_16x16x4_f32(
      false, a, false, bo, (short)0, c, false, false);
  float s = ((d[0] + d[1]) + (d[2] + d[3])) + ((d[4] + d[5]) + (d[6] + d[7]));
  v2f a2; a2[0] = s; a2[1] = 0.0f;
  v8f d2 = __builtin_amdgcn_wmma_f32_16x16x4_f32(
      false, a2, false, bo, (short)0, c, false, false);
  return d2[0];
#else
  #pragma unroll
  for (int off = 16; off > 0; off >>= 1) x += __shfl_xor(x, off, 32);
  return x;
#endif
}

// One block = one (batch b, 8-row phi strip). 256 threads, thread j owns theta
// column j for all 8 rows. Separable Gaussian: a[r][k] (row factor) computed by
// 64 threads into LDS; c[k] (column factor) per thread in registers. Inner loop
// per element: K mul+max for gt, 2 fast logs, ~10 VALU for the focal terms.
__global__ __launch_bounds__(256) void srl_partial(
    const float* __restrict__ pred,
    const float* __restrict__ phi_gt,
    const float* __restrict__ theta_gt,
    const float* __restrict__ grid_phi,
    const float* __restrict__ grid_theta,
    const int*   __restrict__ K_true,
    float*       __restrict__ partial) {
  __shared__ float a_lds[ROWS][KMAX];
  __shared__ float wsum[8];

  const int tid = threadIdx.x;
  const int b   = blockIdx.x / (G_PHI / ROWS);
  const int i0  = (blockIdx.x % (G_PHI / ROWS)) * ROWS;

  const float* prow = pred + ((size_t)b * G_PHI + i0) * G_THETA + tid;
  // Kick HBM fetch of this block's 8KB tile early; overlaps exp + barrier below.
  #pragma unroll
  for (int r = 0; r < ROWS; ++r)
    __builtin_prefetch(prow + (size_t)r * G_THETA, 0, 3);

  // sigma_phi * d_phi = 2*(phi_max-phi_min)/(G_PHI-1); same for theta.
  const float gp0 = grid_phi[0];
  const float gp1 = grid_phi[G_PHI - 1];
  const float gt0 = grid_theta[0];
  const float gt1 = grid_theta[G_THETA - 1];
  const float inv_sp = (float)(G_PHI - 1)   / (2.0f * (gp1 - gp0));
  const float inv_st = (float)(G_THETA - 1) / (2.0f * (gt1 - gt0));
  const int   K = K_true[b];

  // Column factors c[k] (depend on this thread's theta only).
  float c[KMAX];
  const float th = grid_theta[tid];
  #pragma unroll
  for (int k = 0; k < KMAX; ++k) {
    float dd = (th - theta_gt[b * KMAX + k]) * inv_st;
    c[k] = __expf(-0.5f * dd * dd);
  }

  // Row factors a[r][k]: 64 exps per block, shared via LDS.
  if (tid < ROWS * KMAX) {
    int r = tid >> 3, k = tid & 7;
    float dd = (grid_phi[i0 + r] - phi_gt[b * KMAX + k]) * inv_sp;
    a_lds[r][k] = __expf(-0.5f * dd * dd);
  }
  __syncthreads();

  float acc = 0.0f;
  #pragma unroll
  for (int r = 0; r < ROWS; ++r) {
    float p = prow[(size_t)r * G_THETA];
    p = fminf(fmaxf(p, EPS_F), 1.0f - EPS_F);
    float g = 0.0f;
    for (int k = 0; k < K; ++k)                 // K uniform across the block
      g = fmaxf(g, a_lds[r][k] * c[k]);
    const float omp = 1.0f - p;
    const float t1  = F_ALPHA          * omp * omp * __logf(p);    // pos weight
    const float t2  = (1.0f - F_ALPHA) * p   * p   * __logf(omp);  // neg weight
    acc -= t2 + g * (t1 - t2);   // = -(gt*t1 + (1-gt)*t2)
  }

  // Wave reduction via chained WMMAs (total replicated to all lanes),
  // then 8 wave sums combined through LDS.
  float s = wave_sum32(acc);
  if ((tid & 31) == 0) wsum[tid >> 5] = s;
  __syncthreads();
  if (tid == 0) {
    float t = 0.0f;
    #pragma unroll
    for (int w = 0; w < 8; ++w) t += wsum[w];
    partial[blockIdx.x] = t;
  }
}

__global__ __launch_bounds__(256) void srl_finalize(
    const float* __restrict__ partial, float* __restrict__ out) {
  __shared__ float wsum[8];
  float acc = 0.0f;
  for (int i = threadIdx.x; i < NBLK; i += 256) acc += partial[i];
  float s = wave_sum32(acc);
  if ((threadIdx.x & 31) == 0) wsum[threadIdx.x >> 5] = s;
  __syncthreads();
  if (threadIdx.x == 0) {
    float t = 0.0f;
    #pragma unroll
    for (int w = 0; w < 8; ++w) t += wsum[w];
    out[0] = t * (1.0f / ((float)B_DIM * (float)G_PHI * (float)G_THETA));
  }
}

extern "C" void kernel_launch(void* const* d_in, const int* in_sizes, int n_in,
                              void* d_out, int out_size, void* d_ws, size_t ws_size,
                              hipStream_t stream) {
  const float* pred       = (const float*)d_in[0];  // [64,1,512,256]
  const float* phi_gt     = (const float*)d_in[1];  // [64,8]
  const float* theta_gt   = (const float*)d_in[2];  // [64,8]
  const float* grid_phi   = (const float*)d_in[3];  // [512]
  const float* grid_theta = (const float*)d_in[4];  // [256]
  const int*   K_true     = (const int*)  d_in[5];  // [64]
  float* partials = (float*)d_ws;                   // 4096 floats = 16KB scratch

  srl_partial<<<NBLK, 256, 0, stream>>>(pred, phi_gt, theta_gt,
                                        grid_phi, grid_theta, K_true, partials);
  srl_finalize<<<1, 256, 0, stream>>>(partials, (float*)d_out);
}